// RegRankLoss_40699110097504
// MI455X (gfx1250) — compile-verified
//
#include <hip/hip_runtime.h>

// CDNA5 / gfx1250 RankNet pairwise ranking loss.
// - V_WMMA_F32_16X16X4_F32 materializes 16x16 tiles of (p_i - p_j) and
//   (t_i - t_j) directly in WMMA C-layout (rank-2 trick: A=[p_i,1], B=[1;-p_j]).
// - Inputs (64 KB total) staged once per block into LDS; hot loop reads are
//   ds_load broadcasts, lane-masking via branchless v_cndmask.
// - Inner softplus is branchless; raw v_exp_f32/v_log_f32 (base-2) intrinsics
//   since the log argument is in (1,2] (no denorm fixups needed).
// - Symmetry: per-pair term and validity are symmetric in (i,j), so summing
//   the full off-diagonal matrix doubles numerator and denominator; ratio
//   is unchanged vs the reference's upper triangle.

typedef __attribute__((ext_vector_type(2))) float v2f;
typedef __attribute__((ext_vector_type(8))) float v8f;

#define N_ELEM 8192
#define TILE 16
#define NTILES (N_ELEM / TILE)      // 512
#define WAVES_PER_BLOCK 8
#define BLOCK_THREADS (WAVES_PER_BLOCK * 32)

#define LOG2E 1.4426950408889634f
#define LN2   0.6931471805599453f

__global__ void rank_init(double* gsum, unsigned long long* gcnt) {
    if (threadIdx.x == 0) {
        *gsum = 0.0;
        *gcnt = 0ull;
    }
}

__global__ __launch_bounds__(BLOCK_THREADS) void rank_main(
    const float* __restrict__ pred,
    const float* __restrict__ tgt,
    double* __restrict__ gsum,
    unsigned long long* __restrict__ gcnt) {

    // Stage both input vectors into LDS once per block (64 KB of 320 KB/WGP).
    __shared__ float sP[N_ELEM];
    __shared__ float sT[N_ELEM];
    {
        const float4* p4 = (const float4*)pred;
        const float4* t4 = (const float4*)tgt;
        float4* sP4 = (float4*)sP;
        float4* sT4 = (float4*)sT;
        for (int idx = threadIdx.x; idx < N_ELEM / 4; idx += BLOCK_THREADS) {
            sP4[idx] = p4[idx];
            sT4[idx] = t4[idx];
        }
    }
    __syncthreads();

    const int lane  = threadIdx.x & 31;
    const int wave  = threadIdx.x >> 5;
    const int tileI = blockIdx.x;
    const int I0    = tileI * TILE;
    const bool lo   = (lane < 16);
    const int  m0   = (lane >> 4) << 3;  // row base within tile for this half
    const int  nn   = lane & 15;         // column within tile

    // A (16x4): lanes 0-15 hold row M=lane: VGPR0 = K=0 (= p_i / t_i),
    // VGPR1 = K=1 (= 1.0). Lanes 16-31 (K=2,3 rows) = 0.
    const float aOne = lo ? 1.0f : 0.0f;
    const float aP   = lo ? sP[I0 + lane] : 0.0f;
    const float aT   = lo ? sT[I0 + lane] : 0.0f;
    v2f Ap = { aP, aOne };
    v2f At = { aT, aOne };

    float    acc = 0.0f;
    unsigned cnt = 0u;

    for (int tileJ = wave; tileJ < NTILES; tileJ += WAVES_PER_BLOCK) {
        const int J0 = tileJ * TILE;
        // LDS broadcast reads (all 32 lanes read the 16 column values);
        // masking to the B-layout (K=0 row ones, K=1 row -p_j, K=2,3 zero)
        // is a branchless select.
        const float pj = sP[J0 + nn];
        const float tj = sT[J0 + nn];
        v2f Bp = { aOne, lo ? -pj : 0.0f };
        v2f Bt = { aOne, lo ? -tj : 0.0f };

        v8f zero = {};
        // D[m][n] = p_{I0+m} - p_{J0+n}
        v8f dP = __builtin_amdgcn_wmma_f32_16x16x4_f32(
            false, Ap, false, Bp, (short)0, zero, false, false);
        v8f dT = __builtin_amdgcn_wmma_f32_16x16x4_f32(
            false, At, false, Bt, (short)0, zero, false, false);

#pragma unroll
        for (int v = 0; v < 8; ++v) {
            const int   i     = I0 + m0 + v;
            const int   j     = J0 + nn;
            const float td    = dT[v];
            const bool  ok    = (i != j) & (td != 0.0f);
            const float s     = (td > 0.0f) ? 1.0f : -1.0f;
            const float z     = -s * dP[v];                 // BETA = 1
            // stable softplus: max(z,0) + ln(1 + exp(-|z|)), base-2 HW ops.
            const float e     = __builtin_amdgcn_exp2f(-LOG2E * fabsf(z));
            const float sp    = fmaxf(z, 0.0f) + LN2 * __builtin_amdgcn_logf(1.0f + e);
            acc += ok ? sp : 0.0f;
            cnt += ok ? 1u : 0u;
        }
    }

    // Block reduction in LDS, then one global atomic per block.
    __shared__ float    ssum[BLOCK_THREADS];
    __shared__ unsigned scnt[BLOCK_THREADS];
    ssum[threadIdx.x] = acc;
    scnt[threadIdx.x] = cnt;
    __syncthreads();
#pragma unroll
    for (int off = BLOCK_THREADS / 2; off > 0; off >>= 1) {
        if ((int)threadIdx.x < off) {
            ssum[threadIdx.x] += ssum[threadIdx.x + off];
            scnt[threadIdx.x] += scnt[threadIdx.x + off];
        }
        __syncthreads();
    }
    if (threadIdx.x == 0) {
        atomicAdd(gsum, (double)ssum[0]);
        atomicAdd(gcnt, (unsigned long long)scnt[0]);
    }
}

__global__ void rank_final(const double* __restrict__ gsum,
                           const unsigned long long* __restrict__ gcnt,
                           float* __restrict__ out) {
    if (threadIdx.x == 0) {
        const double s = *gsum;
        const unsigned long long c = *gcnt;
        out[0] = (c > 0ull) ? (float)(s / (double)c) : 0.0f;
    }
}

extern "C" void kernel_launch(void* const* d_in, const int* in_sizes, int n_in,
                              void* d_out, int out_size, void* d_ws, size_t ws_size,
                              hipStream_t stream) {
    const float* pred = (const float*)d_in[0];
    const float* tgt  = (const float*)d_in[1];
    double* gsum = (double*)d_ws;
    unsigned long long* gcnt = (unsigned long long*)((char*)d_ws + sizeof(double));

    rank_init<<<1, 32, 0, stream>>>(gsum, gcnt);
    rank_main<<<NTILES, BLOCK_THREADS, 0, stream>>>(pred, tgt, gsum, gcnt);
    rank_final<<<1, 32, 0, stream>>>(gsum, gcnt, (float*)d_out);
}